// LGCN_Encoder_50560355008861
// MI455X (gfx1250) — compile-verified
//
#include <hip/hip_runtime.h>
#include <hip/hip_bf16.h>
#include <stdint.h>

#define N_USERS 100000
#define N_ITEMS 50000
#define N_NODES (N_USERS + N_ITEMS)
#define EMB 64
#define BATCH 4096
#define EDGE_TILE 256

typedef __attribute__((ext_vector_type(2))) float v2f;
typedef __attribute__((ext_vector_type(8))) float v8f;
typedef __attribute__((ext_vector_type(4))) unsigned int u32x4;
typedef __attribute__((ext_vector_type(4))) int i32x4;
typedef __attribute__((ext_vector_type(8))) int i32x8;

#ifndef __has_builtin
#define __has_builtin(x) 0
#endif

#if __has_builtin(__builtin_amdgcn_global_load_async_to_lds_b32)
#define HAVE_ASYNC 1
#else
#define HAVE_ASYNC 0
#endif

#if __has_builtin(__builtin_amdgcn_tensor_load_to_lds)
#define HAVE_TDM 1
#else
#define HAVE_TDM 0
#endif

// Adapts a raw address to WHATEVER pointer type (incl. address-space-qualified)
// a builtin's parameter wants, via a templated user-defined conversion.
struct CastAny {
  uintptr_t v;
  template <typename T>
  __host__ __device__ __forceinline__ operator T() const { return (T)v; }
};

__device__ __forceinline__ void wait_async0() {
#if __has_builtin(__builtin_amdgcn_s_wait_asynccnt)
  __builtin_amdgcn_s_wait_asynccnt(0);
#else
  asm volatile("s_wait_asynccnt 0" ::: "memory");
#endif
}

__device__ __forceinline__ void wait_tensor0() {
#if __has_builtin(__builtin_amdgcn_s_wait_tensorcnt)
  __builtin_amdgcn_s_wait_tensorcnt(0);
#else
  asm volatile("s_wait_tensorcnt 0x0" ::: "memory");
#endif
}

__device__ __forceinline__ void wait_tensor3() {
#if __has_builtin(__builtin_amdgcn_s_wait_tensorcnt)
  __builtin_amdgcn_s_wait_tensorcnt(3);
#else
  asm volatile("s_wait_tensorcnt 0x3" ::: "memory");
#endif
}

#if HAVE_TDM
// 1D tile DMA: stream tile_elems dwords starting at global address `ga`
// into LDS offset `lds_off` via the Tensor Data Mover (ISA 08, D# groups 0/1).
__device__ __forceinline__ void tdm_load_1d(uint32_t lds_off, uint64_t ga,
                                            uint32_t tile_elems,
                                            uint32_t remain_elems) {
  u32x4 g0 = {0u, 0u, 0u, 0u};
  g0.x = 1u;                                        // count=1, user mode
  g0.y = lds_off;                                   // lds_addr (bytes)
  g0.z = (uint32_t)ga;                              // global_addr[31:0]
  g0.w = (uint32_t)((ga >> 32) & 0x01FFFFFFu)       // global_addr[56:32]
         | (2u << 30);                              // type=2 ("image")
  i32x8 g1 = {0, 0, 0, 0, 0, 0, 0, 0};
  g1[0] = (int)(2u << 16);                          // data_size=2 -> 4 bytes
  g1[1] = (int)(remain_elems << 16);                // tensor_dim0[15:0]
  g1[2] = (int)((remain_elems >> 16) & 0xFFFFu)     // tensor_dim0[31:16]
          | (1 << 16);                              // tensor_dim1 = 1
  g1[3] = (int)(tile_elems << 16);                  // tile_dim0
  g1[4] = 1;                                        // tile_dim1 = 1
  g1[5] = (int)remain_elems;                        // tensor_dim0_stride lo
  i32x4 z4 = {0, 0, 0, 0};
#if __clang_major__ >= 23
  i32x8 z8 = {0, 0, 0, 0, 0, 0, 0, 0};
  __builtin_amdgcn_tensor_load_to_lds(g0, g1, z4, z4, z8, 0);
#else
  __builtin_amdgcn_tensor_load_to_lds(g0, g1, z4, z4, 0);
#endif
}
#endif

// ---------------------------------------------------------------- init / zero
__global__ __launch_bounds__(256) void init_kernel(const float* __restrict__ ue,
                                                   const float* __restrict__ ie,
                                                   float* __restrict__ acc,
                                                   float* __restrict__ h) {
  size_t i = (size_t)blockIdx.x * blockDim.x + threadIdx.x;  // float4 index
  const size_t total4 = (size_t)N_NODES * EMB / 4;           // 2.4M
  if (i >= total4) return;
  const size_t u4 = (size_t)N_USERS * EMB / 4;               // 1.6M
  float4 v = (i < u4) ? ((const float4*)ue)[i] : ((const float4*)ie)[i - u4];
  ((float4*)acc)[i] = v;
  ((float4*)h)[i]   = v;
}

__global__ __launch_bounds__(256) void zero_kernel(float4* __restrict__ p, size_t n4) {
  size_t i = (size_t)blockIdx.x * blockDim.x + threadIdx.x;
  if (i < n4) p[i] = make_float4(0.f, 0.f, 0.f, 0.f);
}

// ---------------------------------------------------------------- SpMM (atomic scatter)
// Double-buffered: TDM streams the next 3 edge-tile arrays into LDS while the
// block processes the current tile. 16 lanes per edge: float4 gather of the
// 256B source row (L2-resident), scale, 4x global_atomic_add_f32 scatter.
__global__ __launch_bounds__(256) void spmm_kernel(const int* __restrict__ rows,
                                                   const int* __restrict__ cols,
                                                   const float* __restrict__ vals,
                                                   const float* __restrict__ x,
                                                   float* __restrict__ y,
                                                   int nedges) {
  __shared__ int   sR[2][EDGE_TILE];
  __shared__ int   sC[2][EDGE_TILE];
  __shared__ float sV[2][EDGE_TILE];
  const int t     = threadIdx.x;
  const int group = t >> 4;   // 16 edge-processors per block
  const int sub   = t & 15;   // lane within edge (float4 slot)
  const int tileStride = gridDim.x * EDGE_TILE;
  int base = blockIdx.x * EDGE_TILE;
  if (base >= nedges) return;  // uniform per block

#if HAVE_TDM
  if (t == 0) {
    const uint32_t n0 = (uint32_t)min(EDGE_TILE, nedges - base);
    tdm_load_1d((uint32_t)(uintptr_t)&sR[0][0], (uint64_t)(uintptr_t)(rows + base), n0, (uint32_t)(nedges - base));
    tdm_load_1d((uint32_t)(uintptr_t)&sC[0][0], (uint64_t)(uintptr_t)(cols + base), n0, (uint32_t)(nedges - base));
    tdm_load_1d((uint32_t)(uintptr_t)&sV[0][0], (uint64_t)(uintptr_t)(vals + base), n0, (uint32_t)(nedges - base));
  }
  int buf = 0;
  for (; base < nedges; base += tileStride, buf ^= 1) {
    const int n   = min(EDGE_TILE, nedges - base);
    const int nxt = base + tileStride;
    if (t == 0) {
      if (nxt < nedges) {
        const uint32_t nn = (uint32_t)min(EDGE_TILE, nedges - nxt);
        tdm_load_1d((uint32_t)(uintptr_t)&sR[buf ^ 1][0], (uint64_t)(uintptr_t)(rows + nxt), nn, (uint32_t)(nedges - nxt));
        tdm_load_1d((uint32_t)(uintptr_t)&sC[buf ^ 1][0], (uint64_t)(uintptr_t)(cols + nxt), nn, (uint32_t)(nedges - nxt));
        tdm_load_1d((uint32_t)(uintptr_t)&sV[buf ^ 1][0], (uint64_t)(uintptr_t)(vals + nxt), nn, (uint32_t)(nedges - nxt));
        wait_tensor3();  // in-order: current tile's 3 DMAs are complete
      } else {
        wait_tensor0();
      }
    }
    __syncthreads();
    for (int k = group; k < n; k += 16) {
      const int   r = sR[buf][k];
      const int   c = sC[buf][k];
      const float v = sV[buf][k];
      float4 m = ((const float4*)(x + (size_t)c * EMB))[sub];
      float* yp = y + (size_t)r * EMB + (sub << 2);
      unsafeAtomicAdd(yp + 0, m.x * v);
      unsafeAtomicAdd(yp + 1, m.y * v);
      unsafeAtomicAdd(yp + 2, m.z * v);
      unsafeAtomicAdd(yp + 3, m.w * v);
    }
    __syncthreads();
  }
#else
  for (; base < nedges; base += tileStride) {
    const int n = min(EDGE_TILE, nedges - base);
    if (t < n) {
#if HAVE_ASYNC
      __builtin_amdgcn_global_load_async_to_lds_b32(
          CastAny{(uintptr_t)(rows + base + t)},
          CastAny{(uintptr_t)(uint32_t)(uintptr_t)&sR[0][t]}, 0, 0);
      __builtin_amdgcn_global_load_async_to_lds_b32(
          CastAny{(uintptr_t)(cols + base + t)},
          CastAny{(uintptr_t)(uint32_t)(uintptr_t)&sC[0][t]}, 0, 0);
      __builtin_amdgcn_global_load_async_to_lds_b32(
          CastAny{(uintptr_t)(vals + base + t)},
          CastAny{(uintptr_t)(uint32_t)(uintptr_t)&sV[0][t]}, 0, 0);
#else
      sR[0][t] = rows[base + t];
      sC[0][t] = cols[base + t];
      sV[0][t] = vals[base + t];
#endif
    }
    const int nb = base + tileStride;
    if (nb + t < nedges) {
      __builtin_prefetch(rows + nb + t, 0, 0);
      __builtin_prefetch(cols + nb + t, 0, 0);
      __builtin_prefetch(vals + nb + t, 0, 0);
    }
#if HAVE_ASYNC
    wait_async0();
#endif
    __syncthreads();
    for (int k = group; k < n; k += 16) {
      const int   r = sR[0][k];
      const int   c = sC[0][k];
      const float v = sV[0][k];
      float4 m = ((const float4*)(x + (size_t)c * EMB))[sub];
      float* yp = y + (size_t)r * EMB + (sub << 2);
      unsafeAtomicAdd(yp + 0, m.x * v);
      unsafeAtomicAdd(yp + 1, m.y * v);
      unsafeAtomicAdd(yp + 2, m.z * v);
      unsafeAtomicAdd(yp + 3, m.w * v);
    }
    __syncthreads();
  }
#endif
}

// ---------------------------------------------------------------- acc += h on matrix pipe
// One wave per 16x16 f32 tile. acc_tile = sum_{k=0..3} A_k x B_k + acc_tile,
// where A_k is the K-slice of the 16x16 identity and B_k holds rows 4k..4k+3
// of h  =>  exact elementwise acc += h, via 4 chained V_WMMA_F32_16X16X4_F32.
__global__ __launch_bounds__(256) void acc_add_wmma(const float* __restrict__ h,
                                                    float* __restrict__ acc,
                                                    int totalTiles) {
  const int wave = (int)((blockIdx.x * blockDim.x + threadIdx.x) >> 5);
  if (wave >= totalTiles) return;              // uniform per wave: EXEC stays all-ones
  const int lane = threadIdx.x & 31;
  const int nlo  = lane & 15;
  const int hi   = lane >> 4;

  const int g  = wave >> 2;                    // 16-node group
  const int dt = wave & 3;                     // 16-dim tile
  const int m0 = g * 16;
  const int n0 = dt * 16;
  const size_t base = (size_t)m0 * EMB + n0;

  v8f c;
#pragma unroll
  for (int r = 0; r < 8; ++r)
    c[r] = acc[base + (size_t)(r + hi * 8) * EMB + nlo];

#pragma unroll
  for (int k = 0; k < 4; ++k) {
    // A (16x4): lane holds M=nlo; vgpr0 -> K = hi*2, vgpr1 -> K = hi*2+1
    v2f a;
    a.x = (nlo == 4 * k + hi * 2 + 0) ? 1.0f : 0.0f;
    a.y = (nlo == 4 * k + hi * 2 + 1) ? 1.0f : 0.0f;
    // B (4x16): lane holds N=nlo; vgpr0 -> K = hi*2, vgpr1 -> K = hi*2+1
    v2f b;
    b.x = h[(size_t)(m0 + 4 * k + hi * 2 + 0) * EMB + n0 + nlo];
    b.y = h[(size_t)(m0 + 4 * k + hi * 2 + 1) * EMB + n0 + nlo];
    c = __builtin_amdgcn_wmma_f32_16x16x4_f32(false, a, false, b, (short)0, c,
                                              false, false);
  }

#pragma unroll
  for (int r = 0; r < 8; ++r)
    acc[base + (size_t)(r + hi * 8) * EMB + nlo] = c[r];
}

// ---------------------------------------------------------------- final gather
__global__ __launch_bounds__(256) void gather_kernel(const int* __restrict__ user,
                                                     const int* __restrict__ item,
                                                     const float* __restrict__ acc,
                                                     float* __restrict__ out) {
  const int tid = blockIdx.x * blockDim.x + threadIdx.x;  // float4 granularity
  const int b   = tid >> 4;                               // 0..8191
  const int d4  = tid & 15;
  if (b >= 2 * BATCH) return;
  const int node = (b < BATCH) ? user[b] : (N_USERS + item[b - BATCH]);
  float4 v = ((const float4*)(acc + (size_t)node * EMB))[d4];
  v.x *= 0.25f; v.y *= 0.25f; v.z *= 0.25f; v.w *= 0.25f;
  ((float4*)out)[tid] = v;
}

// ---------------------------------------------------------------- launch
extern "C" void kernel_launch(void* const* d_in, const int* in_sizes, int n_in,
                              void* d_out, int out_size, void* d_ws, size_t ws_size,
                              hipStream_t stream) {
  const int*   user = (const int*)d_in[0];
  const int*   item = (const int*)d_in[1];
  const int*   rows = (const int*)d_in[2];
  const int*   cols = (const int*)d_in[3];
  const float* vals = (const float*)d_in[4];
  const float* uemb = (const float*)d_in[5];
  const float* iemb = (const float*)d_in[6];
  float* out = (float*)d_out;
  const int nedges = in_sizes[2];

  const size_t nElem = (size_t)N_NODES * EMB;  // 9.6M floats
  float* bufA = (float*)d_ws;                  // needs 3*38.4MB of workspace
  float* bufB = bufA + nElem;
  float* acc  = bufB + nElem;

  const int T = 256;
  const size_t n4 = nElem / 4;                 // 2.4M float4
  init_kernel<<<(int)((n4 + T - 1) / T), T, 0, stream>>>(uemb, iemb, acc, bufA);

  float* hc = bufA;
  float* hn = bufB;
  const int tiles = (N_NODES / 16) * (EMB / 16);  // 37500 16x16 tiles
  for (int layer = 0; layer < 3; ++layer) {
    zero_kernel<<<(int)((n4 + T - 1) / T), T, 0, stream>>>((float4*)hn, n4);
    spmm_kernel<<<4096, T, 0, stream>>>(rows, cols, vals, hc, hn, nedges);
    acc_add_wmma<<<(tiles * 32 + T - 1) / T, T, 0, stream>>>(hn, acc, tiles);
    float* tmp = hc; hc = hn; hn = tmp;
  }
  gather_kernel<<<(2 * BATCH * 16 + T - 1) / T, T, 0, stream>>>(user, item, acc, out);
}